// GPPTPrompt_21122649162118
// MI455X (gfx1250) — compile-verified
//
#include <hip/hip_runtime.h>
#include <hip/hip_bf16.h>

#define NN 65536
#define DD 256
#define EE 1048576
#define KK 16
#define CC 40

typedef __attribute__((ext_vector_type(2))) float v2f;
typedef __attribute__((ext_vector_type(8))) float v8f;

// ---------------------------------------------------------------------------
// Kernel 0: zero the accumulator workspace (nb_sum [N*D] + deg [N])
// ---------------------------------------------------------------------------
__global__ void zero_ws_kernel(float* __restrict__ p, size_t n) {
    size_t i = (size_t)blockIdx.x * blockDim.x + threadIdx.x;
    size_t stride = (size_t)gridDim.x * blockDim.x;
    for (; i < n; i += stride) p[i] = 0.0f;
}

// ---------------------------------------------------------------------------
// Kernel 1: edge scatter-add.  One thread per (edge, float4-chunk-of-row).
// h row gather is a coalesced 16B load; accumulator is L2-resident (64 MB),
// so the f32 atomics resolve in L2, not HBM.
// ---------------------------------------------------------------------------
__global__ void scatter_kernel(const float* __restrict__ h,
                               const long long* __restrict__ src,
                               const long long* __restrict__ dst,
                               float* __restrict__ nb_sum,
                               float* __restrict__ deg) {
    unsigned long long tid = (unsigned long long)blockIdx.x * blockDim.x + threadIdx.x;
    unsigned int e = (unsigned int)(tid >> 6);   // 64 float4 chunks per 256-float row
    unsigned int j = (unsigned int)(tid & 63);
    if (e >= EE) return;
    int s = (int)src[e];
    int d = (int)dst[e];
    const float4 v = ((const float4*)(h + (size_t)s * DD))[j];
    float* base = nb_sum + (size_t)d * DD + (size_t)j * 4;
    unsafeAtomicAdd(base + 0, v.x);
    unsafeAtomicAdd(base + 1, v.y);
    unsafeAtomicAdd(base + 2, v.z);
    unsafeAtomicAdd(base + 3, v.w);
    if (j == 0) unsafeAtomicAdd(deg + d, 1.0f);
}

// ---------------------------------------------------------------------------
// Kernel 2: fused  hm = (nb_sum + h)/(deg+1)  ->  logits = hm @ Ws^T (WMMA f32)
//           -> argmax  ->  out = hm @ Wt[idx]^T  (selected expert only).
// One wave32 owns a 16-node tile.  WAVES_PER_BLOCK waves per block.
// ---------------------------------------------------------------------------
#define WAVES_PER_BLOCK 2

__global__ void __launch_bounds__(32 * WAVES_PER_BLOCK)
fused_route_project_kernel(const float* __restrict__ h,
                           const float* __restrict__ nb_sum,
                           const float* __restrict__ deg,
                           const float* __restrict__ Ws,   // [K, D]
                           const float* __restrict__ Wt,   // [K, C, D]
                           float* __restrict__ out) {      // [N, C]
    __shared__ float s_hm[WAVES_PER_BLOCK][16][DD];     // 16 KB per wave
    __shared__ float s_logit[WAVES_PER_BLOCK][16][KK];  // 1 KB per wave
    __shared__ int   s_idx[WAVES_PER_BLOCK][16];

    const int wave = threadIdx.x >> 5;
    const int lane = threadIdx.x & 31;
    const int tile = blockIdx.x * WAVES_PER_BLOCK + wave;
    const int n0 = tile * 16;

    // ---- Phase 1: build hm tile in LDS (16 rows x 256 f32), float4 coalesced
    for (int i = 0; i < 32; ++i) {
        int idx4 = i * 32 + lane;        // 0..1023 float4 elements
        int r    = idx4 >> 6;            // 64 float4 per row
        int c4   = idx4 & 63;
        int n    = n0 + r;
        float4 nb = ((const float4*)(nb_sum + (size_t)n * DD))[c4];
        float4 hh = ((const float4*)(h      + (size_t)n * DD))[c4];
        float inv = 1.0f / (deg[n] + 1.0f);
        float4 hm;
        hm.x = (nb.x + hh.x) * inv;
        hm.y = (nb.y + hh.y) * inv;
        hm.z = (nb.z + hh.z) * inv;
        hm.w = (nb.w + hh.w) * inv;
        ((float4*)(&s_hm[wave][r][0]))[c4] = hm;
    }
    __syncthreads();

    // ---- Phase 2: logits[16x16] = hm_tile[16xD] @ Ws^T[Dx16] via
    //      V_WMMA_F32_16X16X4_F32, K-loop over D in steps of 4.
    // 32-bit A 16x4 layout: lane%16 = M row; lanes<16 hold K={0,1},
    // lanes>=16 hold K={2,3} (VGPR0/VGPR1).  B is the mirrored transpose
    // layout, and since B = Ws^T, B[k][n] = Ws[n][k]: identical addressing.
    const int mrow  = lane & 15;
    const int khalf = (lane >> 4) << 1;   // 0 or 2
    const float* hmrow = &s_hm[wave][mrow][0];
    const float* wsrow = Ws + (size_t)mrow * DD;
    v8f acc = {0.f, 0.f, 0.f, 0.f, 0.f, 0.f, 0.f, 0.f};
    #pragma unroll 4
    for (int d0 = 0; d0 < DD; d0 += 4) {
        v2f a, b;
        a.x = hmrow[d0 + khalf];
        a.y = hmrow[d0 + khalf + 1];
        b.x = wsrow[d0 + khalf];
        b.y = wsrow[d0 + khalf + 1];
        acc = __builtin_amdgcn_wmma_f32_16x16x4_f32(
            /*neg_a=*/false, a, /*neg_b=*/false, b,
            /*c_mod=*/(short)0, acc, /*reuse_a=*/false, /*reuse_b=*/false);
    }

    // C/D layout: VGPR v, lanes 0-15 -> M=v, lanes 16-31 -> M=v+8; N=lane%16
    const int mbase = (lane >> 4) << 3;   // 0 or 8
    #pragma unroll
    for (int v = 0; v < 8; ++v)
        s_logit[wave][mbase + v][lane & 15] = acc[v];
    __syncthreads();

    // ---- Phase 3: per-node argmax over the 16 experts (first-max ties)
    if (lane < 16) {
        float best = s_logit[wave][lane][0];
        int bi = 0;
        #pragma unroll
        for (int k = 1; k < KK; ++k) {
            float val = s_logit[wave][lane][k];
            if (val > best) { best = val; bi = k; }
        }
        s_idx[wave][lane] = bi;
    }
    __syncthreads();

    // ---- Phase 4: selected-expert projection out[n,c] = hm[n] . Wt[idx[n],c]
    // 16 nodes x 40 classes = 640 dot products, 20 per lane.
    // Wt (0.65 MB) is L2/WGP$-resident; hm rows come from LDS as float4.
    for (int p = lane; p < 16 * CC; p += 32) {
        int r = p / CC;
        int c = p - r * CC;
        int k = s_idx[wave][r];
        const float* w   = Wt + ((size_t)(k * CC + c)) * DD;
        const float* hmv = &s_hm[wave][r][0];
        float sum = 0.0f;
        #pragma unroll 8
        for (int d = 0; d < DD; d += 4) {
            float4 wv = *(const float4*)(w + d);
            float4 hv = *(const float4*)(hmv + d);
            sum += wv.x * hv.x + wv.y * hv.y + wv.z * hv.z + wv.w * hv.w;
        }
        out[(size_t)(n0 + r) * CC + c] = sum;
    }
}

// ---------------------------------------------------------------------------
// Launch
// ---------------------------------------------------------------------------
extern "C" void kernel_launch(void* const* d_in, const int* in_sizes, int n_in,
                              void* d_out, int out_size, void* d_ws, size_t ws_size,
                              hipStream_t stream) {
    const float*     h  = (const float*)d_in[0];
    const long long* ei = (const long long*)d_in[1];  // [2, E] int64
    const float*     Ws = (const float*)d_in[2];      // [K, D]
    const float*     Wt = (const float*)d_in[3];      // [K, C, D]
    float*           out = (float*)d_out;             // [N, C]

    float* nb_sum = (float*)d_ws;                     // N*D floats
    float* deg    = nb_sum + (size_t)NN * DD;         // N floats

    // 0) zero accumulators (deterministic each call; ws is poisoned by harness)
    size_t zn = (size_t)NN * DD + NN;
    zero_ws_kernel<<<2048, 256, 0, stream>>>(nb_sum, zn);

    // 1) edge scatter: E * 64 threads (one float4 chunk each)
    {
        unsigned long long total = (unsigned long long)EE * 64ull;
        unsigned int blocks = (unsigned int)(total / 256ull);
        scatter_kernel<<<blocks, 256, 0, stream>>>(h, ei, ei + EE, nb_sum, deg);
    }

    // 2) fused mean + WMMA routing + selected-expert projection
    {
        int tiles = NN / 16;                          // 4096 wave-tiles
        int blocks = tiles / WAVES_PER_BLOCK;         // 2048 blocks of 64 thr
        fused_route_project_kernel<<<blocks, 32 * WAVES_PER_BLOCK, 0, stream>>>(
            h, nb_sum, deg, Ws, Wt, out);
    }
}